// PoseMinkLoc_10746008174742
// MI455X (gfx1250) — compile-verified
//
#include <hip/hip_runtime.h>
#include <hip/hip_bf16.h>
#include <stdint.h>

typedef unsigned long long u64;
typedef unsigned int u32;
typedef float v2f __attribute__((ext_vector_type(2)));
typedef float v8f __attribute__((ext_vector_type(8)));

#define BATCH 32
#define NPTS  262144
#define THREADS 256
#define WARPS 8                      // 256 / wave32
#define KEYS_PER_THREAD 16
#define TILE (THREADS * KEYS_PER_THREAD)      // 4096
#define TILES_PER_BATCH (NPTS / TILE)         // 64
#define RADIX_BITS 11
#define BINS (1 << RADIX_BITS)                // 2048
#define DIGIT_MASK (BINS - 1)
#define BIAS 1024

// ---------------------------------------------------------------------------
// K0: points[B,N,3] f32 -> packed 33-bit keys (x<<22 | y<<11 | z, biased +1024)
// ---------------------------------------------------------------------------
__global__ __launch_bounds__(THREADS) void pack_kernel(
    const float* __restrict__ pts, u64* __restrict__ keys) {
  size_t i = (size_t)blockIdx.x * THREADS + threadIdx.x;
  float x = pts[3 * i + 0];
  float y = pts[3 * i + 1];
  float z = pts[3 * i + 2];
  int cx = (int)floorf(x * 100.0f);   // 1/GRID_SIZE = 100 (exact in f32)
  int cy = (int)floorf(y * 100.0f);
  int cz = (int)floorf(z * 100.0f);
  cx = min(max(cx, -BIAS), BIAS - 1) + BIAS;
  cy = min(max(cy, -BIAS), BIAS - 1) + BIAS;
  cz = min(max(cz, -BIAS), BIAS - 1) + BIAS;
  keys[i] = ((u64)cx << 22) | ((u64)cy << 11) | (u64)cz;
}

// ---------------------------------------------------------------------------
// K1: per-tile digit histogram.  Tile staged into LDS with the CDNA5 async
// global->LDS copy engine (ASYNCcnt), then counted with ds_add atomics.
// hist layout: hist[b][digit][tile]  (tile-contiguous so the scan is linear)
// ---------------------------------------------------------------------------
__global__ __launch_bounds__(THREADS) void hist_kernel(
    const u64* __restrict__ keysIn, u32* __restrict__ hist, int shift) {
  __shared__ __align__(16) u64 stage[TILE];   // 32 KB
  __shared__ u32 cnt[BINS];                   // 8 KB
  const int tile = blockIdx.x, b = blockIdx.y, t = threadIdx.x;

  for (int d = t; d < BINS; d += THREADS) cnt[d] = 0;

  // async copy: 4096 keys = 32KB, 8 x (256 lanes x 16B) transfers
  const u64 gbase =
      (u64)(uintptr_t)(keysIn + (size_t)b * NPTS + (size_t)tile * TILE);
  const u32 lbase = (u32)(size_t)(&stage[0]);   // low 32 bits of flat = LDS offset
#pragma unroll
  for (int k = 0; k < TILE * 8 / (THREADS * 16); ++k) {   // 8 iterations
    u32 off = (u32)(k * THREADS + t) * 16u;
    u64 ga = gbase + off;
    u32 la = lbase + off;
    asm volatile("global_load_async_to_lds_b128 %0, %1, off"
                 :: "v"(la), "v"(ga)
                 : "memory");
  }
  asm volatile("s_wait_asynccnt 0x0" ::: "memory");  // my wave's copies done
  __syncthreads();                                   // all waves' copies done

#pragma unroll
  for (int j = 0; j < KEYS_PER_THREAD; ++j) {
    u64 k = stage[j * THREADS + t];
    u32 d = (u32)(k >> shift) & DIGIT_MASK;
    atomicAdd(&cnt[d], 1u);                          // ds_add_u32
  }
  __syncthreads();

  u32* col = hist + (size_t)b * BINS * TILES_PER_BATCH;
  for (int d = t; d < BINS; d += THREADS)
    col[(size_t)d * TILES_PER_BATCH + tile] = cnt[d];
}

// ---------------------------------------------------------------------------
// K2: one block per batch.  In-place exclusive scan of each digit's tile
// counts, then a block-wide scan of the 256 per-thread totals.
//
// The 256-element block scan is done with the tensor-core-scan identity
//   S = M*U + Lstrict*(M*U*e*1^T)     (U incl. upper-tri ones, Lstrict ones)
// using 8 chained V_WMMA_F32_16X16X4_F32 on wave 0.  All counts are
// integers < 2^24 so every f32 WMMA product/sum is exact.
// ---------------------------------------------------------------------------
__global__ __launch_bounds__(THREADS) void scan_kernel(
    u32* __restrict__ hist, u32* __restrict__ digitBase) {
  const int b = blockIdx.x, t = threadIdx.x;
  const int DPT = BINS / THREADS;   // 8 digits per thread
  u32* hb = hist + (size_t)b * BINS * TILES_PER_BATCH;

  u32 tot[DPT];
  u32 tsum = 0;
  for (int dl = 0; dl < DPT; ++dl) {
    int digit = t * DPT + dl;
    u32* row = hb + (size_t)digit * TILES_PER_BATCH;
    u32 s = 0;
    for (int tile = 0; tile < TILES_PER_BATCH; ++tile) {
      u32 v = row[tile];
      row[tile] = s;                // exclusive prefix within digit (in place)
      s += v;
    }
    tot[dl] = s;
    tsum += s;
  }

  __shared__ float scf[THREADS];
  __shared__ float rowTot[16];
  scf[t] = (float)tsum;
  __syncthreads();

#if __has_builtin(__builtin_amdgcn_wmma_f32_16x16x4_f32)
  if (t < 32) {                     // wave 0, EXEC all ones
    const int l = t;
    const int hi = l >> 4;          // lane half selects K/M upper half
    const int c = l & 15;           // A-row / B-col / C-col index for this lane
    v8f acc = {};                   // C/D: lane l, vgpr j -> (row j+8*hi, col c)

    // rowScan = M x U : 4 chained K=4 WMMAs.
    // A (16x4) lane l, vgpr v -> (row c, K 4k+v+2*hi); B (4x16) -> (K, col c).
#pragma unroll
    for (int k = 0; k < 4; ++k) {
      v2f a, bu;
      a.x = scf[c * 16 + 4 * k + 0 + 2 * hi];
      a.y = scf[c * 16 + 4 * k + 1 + 2 * hi];
      bu.x = (4 * k + 0 + 2 * hi <= c) ? 1.0f : 0.0f;   // U[k'][c]
      bu.y = (4 * k + 1 + 2 * hi <= c) ? 1.0f : 0.0f;
      acc = __builtin_amdgcn_wmma_f32_16x16x4_f32(
          false, a, false, bu, (short)0, acc, false, false);
    }

    // row totals = rowScan[:,15] live in lanes 15 (rows 0-7) and 31 (rows 8-15)
    if (c == 15) {
#pragma unroll
      for (int j = 0; j < 8; ++j) rowTot[j + 8 * hi] = acc[j];
    }
    asm volatile("s_wait_dscnt 0x0" ::: "memory");      // in-wave LDS fence

    // S = rowScan + Lstrict x T, T[k'][c] = rowTot[k'] (broadcast)
#pragma unroll
    for (int k = 0; k < 4; ++k) {
      v2f a2, b2;
      a2.x = (4 * k + 0 + 2 * hi < c) ? 1.0f : 0.0f;    // Lstrict[row=c][k']
      a2.y = (4 * k + 1 + 2 * hi < c) ? 1.0f : 0.0f;
      b2.x = rowTot[4 * k + 0 + 2 * hi];
      b2.y = rowTot[4 * k + 1 + 2 * hi];
      acc = __builtin_amdgcn_wmma_f32_16x16x4_f32(
          false, a2, false, b2, (short)0, acc, false, false);
    }

    // write inclusive scan back (row-major == thread order)
#pragma unroll
    for (int j = 0; j < 8; ++j)
      scf[(j + 8 * hi) * 16 + c] = acc[j];
  }
  __syncthreads();
  u32 run = (u32)scf[t] - tsum;     // exclusive over threads (exact)
#else
  // fallback: Hillis-Steele over 256 totals
  for (int off = 1; off < THREADS; off <<= 1) {
    float v = (t >= off) ? scf[t - off] : 0.0f;
    __syncthreads();
    scf[t] += v;
    __syncthreads();
  }
  u32 run = (u32)scf[t] - tsum;
#endif

  u32* dbb = digitBase + (size_t)b * BINS;
  for (int dl = 0; dl < DPT; ++dl) {
    dbb[t * DPT + dl] = run;
    run += tot[dl];
  }
}

// ---------------------------------------------------------------------------
// K3: stable ranked scatter.  Memory order within tile = (warp, j, lane);
// ranks assigned in the same order: per-warp running LDS counters + wave32
// ballot digit-match + cross-warp exclusive scan.  72 KB LDS / workgroup.
// ---------------------------------------------------------------------------
__global__ __launch_bounds__(THREADS) void scatter_kernel(
    const u64* __restrict__ keysIn, u64* __restrict__ keysOut,
    const u32* __restrict__ tilePrefix, const u32* __restrict__ digitBase,
    int shift) {
  __shared__ u32 warpHist[WARPS * BINS];   // 64 KB (fits 320KB-WGP LDS easily)
  __shared__ u32 ldsOff[BINS];             // 8 KB: digitBase + tilePrefix
  const int tile = blockIdx.x, b = blockIdx.y, t = threadIdx.x;
  const int w = t >> 5, lane = t & 31;
  const u32 lt = (1u << lane) - 1u;

  const u32* tp = tilePrefix + (size_t)b * BINS * TILES_PER_BATCH;
  const u32* db = digitBase + (size_t)b * BINS;
  for (int d = t; d < BINS; d += THREADS)
    ldsOff[d] = db[d] + tp[(size_t)d * TILES_PER_BATCH + tile];
  for (int i = t; i < WARPS * BINS; i += THREADS) warpHist[i] = 0;
  __syncthreads();

  const u64* kin = keysIn + (size_t)b * NPTS + (size_t)tile * TILE +
                   (size_t)w * (32 * KEYS_PER_THREAD) + lane;
  __builtin_prefetch(kin, 0, 0);                       // global_prefetch_b8
  __builtin_prefetch(kin + 32 * (KEYS_PER_THREAD / 2), 0, 0);

  u64 key[KEYS_PER_THREAD];
  u32 dl[KEYS_PER_THREAD];          // digit | (localRank << 16)
  u32* wh = warpHist + w * BINS;
#pragma unroll
  for (int j = 0; j < KEYS_PER_THREAD; ++j) {
    u64 k = kin[j * 32];            // coalesced 256B per wave
    key[j] = k;
    u32 d = (u32)(k >> shift) & DIGIT_MASK;
    // wave32 digit match: 11 ballots
    u32 m = 0xFFFFFFFFu;
#pragma unroll
    for (int bit = 0; bit < RADIX_BITS; ++bit) {
      u32 bb = (u32)__ballot((int)((d >> bit) & 1u));
      m &= ((d >> bit) & 1u) ? bb : ~bb;
    }
    u32 p = __popc(m & lt);         // position among equal-digit lanes
    u32 cntv = wh[d];               // all group lanes broadcast-read
    dl[j] = d | ((cntv + p) << 16);
    if (p == 0) wh[d] = cntv + __popc(m);   // group leader bumps counter
  }
  __syncthreads();

  // exclusive scan over warps per digit, in place
  for (int d = t; d < BINS; d += THREADS) {
    u32 run = 0;
#pragma unroll
    for (int ww = 0; ww < WARPS; ++ww) {
      u32 v = warpHist[ww * BINS + d];
      warpHist[ww * BINS + d] = run;
      run += v;
    }
  }
  __syncthreads();

  u64* kout = keysOut + (size_t)b * NPTS;
#pragma unroll
  for (int j = 0; j < KEYS_PER_THREAD; ++j) {
    u32 d = dl[j] & DIGIT_MASK;
    u32 rank = ldsOff[d] + wh[d] + (dl[j] >> 16);
    kout[rank] = key[j];            // scattered 8B writes -> absorbed by 192MB L2
  }
}

// ---------------------------------------------------------------------------
// K4: zero counts section (d_out is poisoned; must re-zero every call)
// ---------------------------------------------------------------------------
__global__ void zero_counts(float* __restrict__ out) {
  out[(size_t)BATCH * NPTS * 5 + threadIdx.x] = 0.0f;
}

// ---------------------------------------------------------------------------
// K5: unpack sorted keys -> coords [B,N,4], features [B,N,1], counts [B]
// ---------------------------------------------------------------------------
__global__ __launch_bounds__(THREADS) void unpack_kernel(
    const u64* __restrict__ keys, float* __restrict__ out) {
  __shared__ u32 partial;
  if (threadIdx.x == 0) partial = 0;
  __syncthreads();

  size_t i = (size_t)blockIdx.x * THREADS + threadIdx.x;
  int b = (int)(i / NPTS);          // N % THREADS == 0 -> one batch per block
  int n = (int)(i % NPTS);
  u64 k = keys[i];
  int first = (n == 0) ? 1 : (keys[i - 1] != k ? 1 : 0);

  float fx = (float)((int)((k >> 22) & DIGIT_MASK) - BIAS);
  float fy = (float)((int)((k >> 11) & DIGIT_MASK) - BIAS);
  float fz = (float)((int)(k & DIGIT_MASK) - BIAS);
  float* crd = out + i * 4;
  crd[0] = (float)b; crd[1] = fx; crd[2] = fy; crd[3] = fz;
  out[(size_t)BATCH * NPTS * 4 + i] = (float)first;

  atomicAdd(&partial, (u32)first);  // ds_add_u32
  __syncthreads();
  if (threadIdx.x == 0)
    atomicAdd(out + (size_t)BATCH * NPTS * 5 + b, (float)partial);
}

// ---------------------------------------------------------------------------
extern "C" void kernel_launch(void* const* d_in, const int* in_sizes, int n_in,
                              void* d_out, int out_size, void* d_ws,
                              size_t ws_size, hipStream_t stream) {
  const float* pts = (const float*)d_in[0];
  float* out = (float*)d_out;

  // workspace layout (~151 MB)
  u64* keysA = (u64*)d_ws;
  u64* keysB = keysA + (size_t)BATCH * NPTS;
  u32* hist = (u32*)(keysB + (size_t)BATCH * NPTS);
  u32* digitBase = hist + (size_t)BATCH * BINS * TILES_PER_BATCH;

  pack_kernel<<<(BATCH * NPTS) / THREADS, THREADS, 0, stream>>>(pts, keysA);

  u64* cur = keysA;
  u64* nxt = keysB;
  for (int pass = 0; pass < 3; ++pass) {        // 33-bit key, 3 x 11-bit LSD
    int shift = pass * RADIX_BITS;
    hist_kernel<<<dim3(TILES_PER_BATCH, BATCH), THREADS, 0, stream>>>(
        cur, hist, shift);
    scan_kernel<<<BATCH, THREADS, 0, stream>>>(hist, digitBase);
    scatter_kernel<<<dim3(TILES_PER_BATCH, BATCH), THREADS, 0, stream>>>(
        cur, nxt, hist, digitBase, shift);
    u64* tmp = cur; cur = nxt; nxt = tmp;
  }
  // after 3 swaps, sorted keys are in `cur` (== keysB)

  zero_counts<<<1, BATCH, 0, stream>>>(out);
  unpack_kernel<<<(BATCH * NPTS) / THREADS, THREADS, 0, stream>>>(cur, out);
}